// MiniRocketFeaturesPlus_72181220376587
// MI455X (gfx1250) — compile-verified
//
#include <hip/hip_runtime.h>
#include <hip/hip_bf16.h>
#include <cmath>

// ---------------------------------------------------------------------------
// MiniRocket fused kernel for gfx1250 (MI455X).
// Conv expressed as GEMM:  C[84 x 2048] = W_i[84 x 81] * X_{b,i}[81 x 2048]
// with W_i[j, c*9+t] = channel_comb[i,c,j] * kernel[j,t]  (values in {0,-1,2})
// and  X[(c,t), l]   = x[b, c, l + (t-4)*d]  (zero outside [0,2048)).
// GEMM runs on V_WMMA_F32_16X16X4_F32 with the A-panel held in registers and
// x[b] staged in LDS; PPV stats accumulate in registers.
// ---------------------------------------------------------------------------

typedef __attribute__((ext_vector_type(2))) float v2f;
typedef __attribute__((ext_vector_type(8))) float v8f;

#define C_IN        9
#define SEQ_LEN     2048
#define KSIZE       9
#define NKER        84
#define KTOT        81      // C_IN * KSIZE
#define KPAD        84      // 81 padded to 21 K-steps of 4
#define KSTEPS      (KPAD / 4)
#define MPAD        96      // 84 kernels padded to 6 WMMA M-tiles
#define NWAVE       6
#define BLK         (NWAVE * 32)
#define MAXD        32
#define ITEMS       4       // ceil(84 * max_nf / BLK)

#define XLDS_FLOATS (C_IN * SEQ_LEN)                       // 18432 floats = 73728 B
#define CT_OFF      (XLDS_FLOATS * 4)                      // 73728
#define KTAB_OFF    (CT_OFF + MPAD * 33 * 4)               // +12672 = 86400
#define SUM_OFF     (KTAB_OFF + KPAD * 8)                  // +672   = 87072
#define SMEM_BYTES  (SUM_OFF + NKER * 4)                   // +336   = 87408

struct DilCfg {
    int D;          // number of unique dilations (~26)
    int Ftot;       // total feature columns (9912)
    int nfsum;      // features per kernel (59) == bias row stride
    int dil[MAXD];
    int nf[MAXD];
    int base[MAXD];   // output column base per dilation
    int nfoff[MAXD];  // bias column offset per dilation
};

// Build padded per-dilation weight matrices W[i][row(96)][k(84)].
__global__ void mr_build_w(const float* __restrict__ ker,   // (756,1,9): ker[j*9+t] for j<84
                           const float* __restrict__ cc,    // (D,9,84)
                           float* __restrict__ W, int D)
{
    int i = blockIdx.x;
    if (i >= D) return;
    for (int idx = threadIdx.x; idx < MPAD * KPAD; idx += blockDim.x) {
        int row = idx / KPAD;
        int k   = idx - row * KPAD;
        float v = 0.f;
        if (row < NKER && k < KTOT) {
            int c = k / KSIZE;
            int t = k - c * KSIZE;
            v = cc[(size_t)i * C_IN * NKER + c * NKER + row] * ker[row * KSIZE + t];
        }
        W[(size_t)i * MPAD * KPAD + idx] = v;
    }
}

__global__ __launch_bounds__(BLK)
void mr_main(const float* __restrict__ x,       // (B, 9, 2048)
             const float* __restrict__ W,       // (D, 96, 84)
             const float* __restrict__ biases,  // (84, nfsum)
             float* __restrict__ out,           // (B, Ftot)
             DilCfg cfg)
{
    const int b  = blockIdx.x;
    const int i  = blockIdx.y;
    const int d  = cfg.dil[i];
    const int nf = cfg.nf[i];
    const int p  = 4 * d;
    const int p1 = i & 1;

    const float* xb = x + (size_t)b * C_IN * SEQ_LEN;
    const float* Wi = W + (size_t)i * MPAD * KPAD;

    extern __shared__ unsigned char smemRaw[];
    float* xlds        = (float*)smemRaw;                         // 9 x 2048 f32
    float (*Ct)[33]    = (float(*)[33])(smemRaw + CT_OFF);        // [MPAD][33]
    int2*  kTab        = (int2*)(smemRaw + KTAB_OFF);             // {shift, chanBase}
    float* sumLds      = (float*)(smemRaw + SUM_OFF);             // [NKER]

    const int tid  = threadIdx.x;
    const int lane = tid & 31;
    const int wv   = tid >> 5;       // 0..5
    const int m0   = wv * 16;
    const int lrow = lane & 15;
    const int lhi  = lane >> 4;      // 0 or 1 (half-wave)

    // ---- stage x[b] into LDS (vectorized) ----------------------------------
    {
        const float4* xs = (const float4*)xb;
        float4*       xd = (float4*)xlds;
        for (int idx = tid; idx < XLDS_FLOATS / 4; idx += BLK) xd[idx] = xs[idx];
    }

    // ---- per-k lookup table: shift and channel base ------------------------
    if (tid < KPAD) {
        int2 e;
        if (tid < KTOT) {
            int c = tid / KSIZE;
            int t = tid - c * KSIZE;
            e.x = (t - 4) * d;
            e.y = c * SEQ_LEN;
        } else {
            e.x = -(1 << 28);        // forces position out-of-range -> selects 0
            e.y = 0;
        }
        kTab[tid] = e;
    }

    // ---- per-thread PPV items: (kernel m, feature f) pairs -----------------
    int   itm_m[ITEMS], itm_lo[ITEMS], itm_hi[ITEMS];
    float itm_bias[ITEMS], itm_cnt[ITEMS], itm_abs[ITEMS];
    const int nitems = NKER * nf;
#pragma unroll
    for (int s = 0; s < ITEMS; ++s) {
        int it = tid + s * BLK;
        if (it < nitems) {
            int m = it / nf;
            int f = it - m * nf;
            itm_m[s] = m;
            int lo = ((m & 1) == p1) ? 0 : p;   // cropped range for other parity
            itm_lo[s] = lo;
            itm_hi[s] = SEQ_LEN - lo;
            itm_bias[s] = biases[m * cfg.nfsum + cfg.nfoff[i] + f];
            itm_cnt[s] = 0.f;
            itm_abs[s] = 0.f;
        } else {
            itm_m[s] = -1; itm_lo[s] = 0; itm_hi[s] = 0;
            itm_bias[s] = 0.f; itm_cnt[s] = 0.f; itm_abs[s] = 0.f;
        }
    }
    // per-kernel running sum of C over that kernel's valid range
    float mySum = 0.f;
    const int myM = (tid < NKER) ? tid : -1;
    int myLo = 0, myHi = 0;
    if (myM >= 0) {
        myLo = ((myM & 1) == p1) ? 0 : p;
        myHi = SEQ_LEN - myLo;
    }

    // ---- A panel: resident in registers for the whole kernel ---------------
    v2f areg[KSTEPS];
#pragma unroll
    for (int ks = 0; ks < KSTEPS; ++ks)
        areg[ks] = *(const v2f*)(Wi + (m0 + lrow) * KPAD + 4 * ks + (lhi << 1));

    __syncthreads();

    // ---- main loop: 64 position tiles of 32 --------------------------------
    for (int tile = 0; tile < SEQ_LEN / 32; ++tile) {
        const int l0 = tile * 32;
        v8f acc0 = {};   // positions l0 + 0..15
        v8f acc1 = {};   // positions l0 + 16..31

#pragma unroll
        for (int ks = 0; ks < KSTEPS; ++ks) {     // fully unrolled: areg[] const-indexed
            const int kb = 4 * ks + (lhi << 1);
            // B fragments (4x16 f32): lane holds X[kb+q][l0(+16)+lrow]
            v2f b0, b1;
#pragma unroll
            for (int q = 0; q < 2; ++q) {
                const int2 e = kTab[kb + q];
                const int la = l0 + lrow + e.x;   // position, N-tile 0
                const int lb = la + 16;           // position, N-tile 1
                const int lac = min(max(la, 0), SEQ_LEN - 1);   // v_med3
                const int lbc = min(max(lb, 0), SEQ_LEN - 1);
                float va = xlds[e.y + lac];                      // LDS, always in-range
                float vb = xlds[e.y + lbc];
                va = ((unsigned)la < (unsigned)SEQ_LEN) ? va : 0.f;
                vb = ((unsigned)lb < (unsigned)SEQ_LEN) ? vb : 0.f;
                if (q == 0) { b0.x = va; b1.x = vb; }
                else        { b0.y = va; b1.y = vb; }
            }
            acc0 = __builtin_amdgcn_wmma_f32_16x16x4_f32(
                       false, areg[ks], false, b0, (short)0, acc0, false, false);
            acc1 = __builtin_amdgcn_wmma_f32_16x16x4_f32(
                       false, areg[ks], false, b1, (short)0, acc1, false, false);
        }

        // ---- spill C tile to LDS (D-matrix layout -> [m][pos]) -------------
#pragma unroll
        for (int r = 0; r < 8; ++r) {
            const int m = m0 + r + 8 * lhi;
            Ct[m][lrow]      = acc0[r];
            Ct[m][lrow + 16] = acc1[r];
        }
        __syncthreads();

        // ---- PPV accumulation over this tile's 32 positions ----------------
#pragma unroll
        for (int s = 0; s < ITEMS; ++s) {
            if (itm_m[s] >= 0) {
                const int   m  = itm_m[s];
                const float bv = itm_bias[s];
                const int   lo = itm_lo[s], hi = itm_hi[s];
                float cnt = itm_cnt[s], ab = itm_abs[s];
                for (int pp = 0; pp < 32; ++pp) {
                    const int l = l0 + pp;
                    const float inr = (l >= lo && l < hi) ? 1.f : 0.f;
                    const float dv  = Ct[m][pp] - bv;
                    cnt += (dv > 0.f) ? inr : 0.f;
                    ab   = fmaf(inr, fabsf(dv), ab);
                }
                itm_cnt[s] = cnt; itm_abs[s] = ab;
            }
        }
        if (myM >= 0) {
            float sm = mySum;
            for (int pp = 0; pp < 32; ++pp) {
                const int l = l0 + pp;
                const float inr = (l >= myLo && l < myHi) ? 1.f : 0.f;
                sm = fmaf(inr, Ct[myM][pp], sm);
            }
            mySum = sm;
        }
        __syncthreads();
    }

    if (myM >= 0) sumLds[myM] = mySum;
    __syncthreads();

    // ---- finalize features and write output --------------------------------
    // Layout per dilation: [a(par p1) | b(par p1) | a(other, cropped) | b(other)],
    // each block 42*nf wide; kernel m -> slot (m>>1), feature f.
    float* orow = out + (size_t)b * cfg.Ftot + cfg.base[i];
#pragma unroll
    for (int s = 0; s < ITEMS; ++s) {
        if (itm_m[s] < 0) continue;
        const int m  = itm_m[s];
        const int it = tid + s * BLK;
        const int f  = it - m * nf;
        const float Lm = (float)(itm_hi[s] - itm_lo[s]);
        const float aFeat  = itm_cnt[s] / Lm;
        const float sumdif = sumLds[m] - Lm * itm_bias[s];
        const float reluS  = 0.5f * (sumdif + itm_abs[s]);
        const float bFeat  = reluS / fmaxf(itm_abs[s], 1e-8f);
        const int sidx = m >> 1;
        const int grp  = ((m & 1) == p1) ? 0 : 2;
        orow[(grp    ) * 42 * nf + sidx * nf + f] = aFeat;
        orow[(grp + 1) * 42 * nf + sidx * nf + f] = bFeat;
    }
}

// ---------------------------------------------------------------------------
// Host side
// ---------------------------------------------------------------------------
static DilCfg make_config()
{
    DilCfg cfg{};
    const int nfpk     = (10000 / 2) / NKER;                 // 59
    const int true_max = (nfpk < 32) ? nfpk : 32;            // 32
    const double multiplier = (double)nfpk / (double)true_max;
    const double max_exp = log2((double)(SEQ_LEN - 1) / (double)(KSIZE - 1));
    const double step = max_exp / (double)(true_max - 1);

    int raw[MAXD];
    for (int q = 0; q < true_max; ++q) {
        double e = (q == true_max - 1) ? max_exp : (double)q * step;  // numpy linspace
        raw[q] = (int)exp2(e);                                        // trunc, as astype(int32)
    }
    // unique + counts (raw is nondecreasing)
    int D = 0, cnts[MAXD];
    for (int q = 0; q < true_max; ++q) {
        if (D > 0 && raw[q] == cfg.dil[D - 1]) cnts[D - 1]++;
        else { cfg.dil[D] = raw[q]; cnts[D] = 1; D++; }
    }
    cfg.D = D;
    int ssum = 0;
    for (int u = 0; u < D; ++u) { cfg.nf[u] = (int)((double)cnts[u] * multiplier); ssum += cfg.nf[u]; }
    int rem = nfpk - ssum, ii = 0;
    while (rem > 0) { cfg.nf[ii]++; rem--; ii = (ii + 1) % D; }

    int cbase = 0, foff = 0;
    for (int u = 0; u < D; ++u) {
        cfg.base[u]  = cbase;
        cfg.nfoff[u] = foff;
        cbase += 168 * cfg.nf[u];   // 4 groups x 42 kernels x nf
        foff  += cfg.nf[u];
    }
    cfg.Ftot  = cbase;   // 9912
    cfg.nfsum = foff;    // 59
    return cfg;
}

extern "C" void kernel_launch(void* const* d_in, const int* in_sizes, int n_in,
                              void* d_out, int out_size, void* d_ws, size_t ws_size,
                              hipStream_t stream)
{
    (void)n_in; (void)out_size; (void)ws_size;
    const float* x       = (const float*)d_in[0];
    const float* kernels = (const float*)d_in[1];
    const float* cc      = (const float*)d_in[2];
    const float* biases  = (const float*)d_in[3];
    float*       out     = (float*)d_out;
    float*       Wpad    = (float*)d_ws;          // D * 96 * 84 f32 (~840 KB)

    const DilCfg cfg = make_config();
    const int B = in_sizes[0] / (C_IN * SEQ_LEN); // 32

    // allow > default dynamic LDS (87.4 KB; WGP has 320 KB). Not a stream op,
    // graph-capture safe, deterministic.
    (void)hipFuncSetAttribute((const void*)mr_main,
                              hipFuncAttributeMaxDynamicSharedMemorySize,
                              SMEM_BYTES);

    mr_build_w<<<cfg.D, 256, 0, stream>>>(kernels, cc, Wpad, cfg.D);

    dim3 grid(B, cfg.D);
    mr_main<<<grid, BLK, SMEM_BYTES, stream>>>(x, Wpad, biases, out, cfg);
}